// CRFCell_33517924778639
// MI455X (gfx1250) — compile-verified
//
#include <hip/hip_runtime.h>

#define NT  24
#define SEQ 1024
#define NB  1024

#define L2E 1.44269504088896340736f   /* log2(e) */
#define LN2 0.69314718055994530942f   /* ln(2)   */

typedef float v2f __attribute__((ext_vector_type(2)));
typedef float v8f __attribute__((ext_vector_type(8)));

__device__ __forceinline__ float xor16f(float v, int m) { return __shfl_xor(v, m, 32); }

// raw hardware transcendentals (base-2), no denormal fixup sequences
__device__ __forceinline__ float exp2_raw(float x)
{
#if defined(__HIP_DEVICE_COMPILE__) && __has_builtin(__builtin_amdgcn_exp2f)
    return __builtin_amdgcn_exp2f(x);
#else
    return exp2f(x);
#endif
}
__device__ __forceinline__ float log2_raw(float x)
{
#if defined(__HIP_DEVICE_COMPILE__) && __has_builtin(__builtin_amdgcn_logf)
    return __builtin_amdgcn_logf(x);
#else
    return log2f(x);
#endif
}

// exact branchless select: cond ? a : b  (forces both operands to be computed)
__device__ __forceinline__ float bitsel(unsigned msk, float a, float b)
{
    return __uint_as_float((__float_as_uint(a) & msk) | (__float_as_uint(b) & ~msk));
}

// ---------------------------------------------------------------------------
// Forward-algorithm scan (denominator). One wave handles 16 batch rows.
// 16 waves per workgroup -> 4 waves per SIMD32 so independent scan chains
// interleave and hide the serial WMMA->log->exp dependency latency.
// State kept in log2 domain:  sc2 = score * log2(e).
// Step:  sc2'_j = m2 + log2( sum_i 2^(sc2_i - m2) * e^T[i,j] ) + em_j*log2(e)
// The matvec runs on V_WMMA_F32_16X16X4_F32:
//   A = (e^T)^T   (constant, 2 out-tag halves x 6 K-chunks of 16x4)
//   B = p         (probability state, 4x16 chunks, batch along lanes)
//   D = s         (tags along VGPRs, batch along lanes)
// Slot layout per lane (batch b = lane&15):
//   lane b    : slots 0..7 -> tags 0..7 ; slots 8..15 -> tags 16..23
//   lane b+16 : slots 0..7 -> tags 8..15; slots 8..15 unused
// ---------------------------------------------------------------------------
__global__ __launch_bounds__(512)
void crf_forward_scan(const float* __restrict__ feat,
                      const int*   __restrict__ mask,
                      const float* __restrict__ startT,
                      const float* __restrict__ endT,
                      const float* __restrict__ trans,
                      float* __restrict__ out)
{
    const int  lane  = threadIdx.x & 31;
    const int  wave  = threadIdx.x >> 5;
    const int  group = blockIdx.x * 16 + wave;       // 64 groups of 16 batches
    const bool lo    = lane < 16;
    const int  bb    = group * 16 + (lane & 15);
    const int  t0off = lo ? 0  : 8;   // tags for slots 0..7
    const int  t1off = lo ? 16 : 8;   // tags for slots 8..15 (hi: dup, unused)

    // ---- constant A tiles: A[h][c][row j'][k'] = exp(T[4c + k' , 16h + j'])
    v2f A0[6], A1[6];
    {
        const int r  = lane & 15;
        const int kk = lo ? 0 : 2;
#pragma unroll
        for (int c = 0; c < 6; ++c) {
            const int i0 = 4 * c + kk, i1 = i0 + 1;
            A0[c].x = __expf(trans[i0 * NT + r]);
            A0[c].y = __expf(trans[i1 * NT + r]);
            const int j1 = 16 + (r < 8 ? r : 7);          // clamp: stay in-bounds
            const float e0 = __expf(trans[i0 * NT + j1]);
            const float e1 = __expf(trans[i1 * NT + j1]);
            A1[c].x = (r < 8) ? e0 : 0.0f;                 // zero-pad rows 24..31
            A1[c].y = (r < 8) ? e1 : 0.0f;
        }
    }

    // ---- initial state (log2 domain): sc2 = (start_transitions + em[0]) * log2e
    const float* frow = feat + (size_t)bb * SEQ * NT;
    float sc[16];
    {
        const float4 e0 = *(const float4*)(frow + t0off);
        const float4 e1 = *(const float4*)(frow + t0off + 4);
        const float4 e2 = *(const float4*)(frow + t1off);
        const float4 e3 = *(const float4*)(frow + t1off + 4);
        const float4 s0 = *(const float4*)(startT + t0off);
        const float4 s1 = *(const float4*)(startT + t0off + 4);
        const float4 s2 = *(const float4*)(startT + t1off);
        const float4 s3 = *(const float4*)(startT + t1off + 4);
        sc[0]=(e0.x+s0.x)*L2E; sc[1]=(e0.y+s0.y)*L2E; sc[2]=(e0.z+s0.z)*L2E; sc[3]=(e0.w+s0.w)*L2E;
        sc[4]=(e1.x+s1.x)*L2E; sc[5]=(e1.y+s1.y)*L2E; sc[6]=(e1.z+s1.z)*L2E; sc[7]=(e1.w+s1.w)*L2E;
        sc[8]=(e2.x+s2.x)*L2E; sc[9]=(e2.y+s2.y)*L2E; sc[10]=(e2.z+s2.z)*L2E; sc[11]=(e2.w+s2.w)*L2E;
        sc[12]=(e3.x+s3.x)*L2E; sc[13]=(e3.y+s3.y)*L2E; sc[14]=(e3.z+s3.z)*L2E; sc[15]=(e3.w+s3.w)*L2E;
    }

    float mPrev;
    v2f   Bc[6];
    // repack: sc2 -> (row max m2, p = 2^(sc2-m2), B chunks via xor-16 swaps)
    auto repack = [&]() {
        float lm = sc[0];
#pragma unroll
        for (int s = 1; s < 8; ++s) lm = fmaxf(lm, sc[s]);
        float lm2 = sc[8];
#pragma unroll
        for (int s = 9; s < 16; ++s) lm2 = fmaxf(lm2, sc[s]);
        if (lo) lm = fmaxf(lm, lm2);                 // hi lanes: only 8 valid slots
        const float m = fmaxf(lm, xor16f(lm, 16));   // full 24-tag max per batch
        float p[16], pp[16];
#pragma unroll
        for (int s = 0; s < 16; ++s) p[s] = exp2_raw(sc[s] - m);
#pragma unroll
        for (int s = 0; s < 16; ++s) pp[s] = xor16f(p[s], 16);
        // B chunk c: x = p[tag 4c+(lo?0:2)], y = p[tag 4c+(lo?1:3)], col = lane&15
        Bc[0].x = lo ? p[0]  : pp[2];   Bc[0].y = lo ? p[1]  : pp[3];
        Bc[1].x = lo ? p[4]  : pp[6];   Bc[1].y = lo ? p[5]  : pp[7];
        Bc[2].x = lo ? pp[0] : p[2];    Bc[2].y = lo ? pp[1] : p[3];
        Bc[3].x = lo ? pp[4] : p[6];    Bc[3].y = lo ? pp[5] : p[7];
        Bc[4].x = lo ? p[8]  : pp[10];  Bc[4].y = lo ? p[9]  : pp[11];
        Bc[5].x = lo ? p[12] : pp[14];  Bc[5].y = lo ? p[13] : pp[15];
        mPrev = m;
    };
    repack();

    const int* mrow = mask + (size_t)bb * SEQ;

    for (int t = 1; t < SEQ; ++t) {
        const float* fr = frow + (size_t)t * NT;
        const float4 e0 = *(const float4*)(fr + t0off);
        const float4 e1 = *(const float4*)(fr + t0off + 4);
        const float4 e2 = *(const float4*)(fr + t1off);
        const float4 e3 = *(const float4*)(fr + t1off + 4);
        const unsigned msk = mrow[t] ? 0xFFFFFFFFu : 0u;

        // unconditional prefetch of the row consumed 8 steps from now (into GL2)
        const int tp = (t + 8 < SEQ) ? (t + 8) : (SEQ - 1);
        __builtin_prefetch(frow + (size_t)tp * NT, 0, 3);

        v8f D0 = {0.f,0.f,0.f,0.f,0.f,0.f,0.f,0.f};
        v8f D1 = {0.f,0.f,0.f,0.f,0.f,0.f,0.f,0.f};
#pragma unroll
        for (int c = 0; c < 6; ++c) {
            D0 = __builtin_amdgcn_wmma_f32_16x16x4_f32(false, A0[c], false, Bc[c],
                                                       (short)0, D0, false, false);
            D1 = __builtin_amdgcn_wmma_f32_16x16x4_f32(false, A1[c], false, Bc[c],
                                                       (short)0, D1, false, false);
        }

        const float em[16] = { e0.x,e0.y,e0.z,e0.w, e1.x,e1.y,e1.z,e1.w,
                               e2.x,e2.y,e2.z,e2.w, e3.x,e3.y,e3.z,e3.w };
#pragma unroll
        for (int s = 0; s < 16; ++s) {
            const float sv = (s < 8) ? D0[s] : D1[s - 8];
            // ns2 = m2 + log2(sv) + em*log2e  (em scale folded into the fma)
            const float ns = fmaf(em[s], L2E, mPrev) + log2_raw(sv);
            sc[s] = bitsel(msk, ns, sc[s]);    // exact, branchless masked update
        }
        repack();
    }

    // ---- finalize: denom_b = ln2 * log2-sum-exp2_j(sc2_j + endT_j*log2e)
    {
        const float4 q0 = *(const float4*)(endT + t0off);
        const float4 q1 = *(const float4*)(endT + t0off + 4);
        const float4 q2 = *(const float4*)(endT + t1off);
        const float4 q3 = *(const float4*)(endT + t1off + 4);
        const float et[16] = { q0.x,q0.y,q0.z,q0.w, q1.x,q1.y,q1.z,q1.w,
                               q2.x,q2.y,q2.z,q2.w, q3.x,q3.y,q3.z,q3.w };
#pragma unroll
        for (int s = 0; s < 16; ++s) sc[s] = fmaf(et[s], L2E, sc[s]);

        float lm = sc[0];
#pragma unroll
        for (int s = 1; s < 8; ++s) lm = fmaxf(lm, sc[s]);
        float lm2 = sc[8];
#pragma unroll
        for (int s = 9; s < 16; ++s) lm2 = fmaxf(lm2, sc[s]);
        if (lo) lm = fmaxf(lm, lm2);
        const float m = fmaxf(lm, xor16f(lm, 16));

        float sum = 0.f, sum2 = 0.f;
#pragma unroll
        for (int s = 0; s < 8; ++s)  sum  += exp2_raw(sc[s] - m);
#pragma unroll
        for (int s = 8; s < 16; ++s) sum2 += exp2_raw(sc[s] - m);
        if (lo) sum += sum2;
        sum += xor16f(sum, 16);
        const float denom = (m + log2_raw(sum)) * LN2;   // back to natural log

        float contrib = lo ? denom : 0.0f;
#pragma unroll
        for (int w = 16; w >= 1; w >>= 1) contrib += xor16f(contrib, w);
        if (lane == 0) atomicAdd(out, contrib * (1.0f / NB));
    }
}

// ---------------------------------------------------------------------------
// Numerator: gold-path score, one wave per batch row, lanes strided over t.
// 8 waves per block.
// ---------------------------------------------------------------------------
__global__ __launch_bounds__(256)
void crf_numerator(const float* __restrict__ feat,
                   const int*   __restrict__ mask,
                   const int*   __restrict__ target,
                   const float* __restrict__ startT,
                   const float* __restrict__ endT,
                   const float* __restrict__ trans,
                   float* __restrict__ out)
{
    const int lane = threadIdx.x & 31;
    const int b    = blockIdx.x * 8 + (threadIdx.x >> 5);
    const int*   tg = target + (size_t)b * SEQ;
    const int*   mk = mask   + (size_t)b * SEQ;
    const float* em = feat   + (size_t)b * SEQ * NT;

    float acc = 0.f;
    int   msum = 0;
    for (int t = lane; t < SEQ; t += 32) msum += mk[t];
    for (int t = 1 + lane; t < SEQ; t += 32) {
        const int ct = tg[t], pt = tg[t - 1];
        const float v = trans[pt * NT + ct] + em[(size_t)t * NT + ct];
        const unsigned msk = mk[t] ? 0xFFFFFFFFu : 0u;
        acc += bitsel(msk, v, 0.0f);
    }
#pragma unroll
    for (int w = 16; w >= 1; w >>= 1) {
        acc  += xor16f(acc, w);
        msum += __shfl_xor(msum, w, 32);
    }
    if (lane == 0) {
        const int tag0 = tg[0];
        const int lastTag = tg[msum - 1];
        const float num = acc + startT[tag0] + em[tag0] + endT[lastTag];
        atomicAdd(out, -num * (1.0f / NB));
    }
}

__global__ void zero_out(float* out, int n)
{
    const int i = threadIdx.x;
    if (i < n) out[i] = 0.0f;
}

extern "C" void kernel_launch(void* const* d_in, const int* in_sizes, int n_in,
                              void* d_out, int out_size, void* d_ws, size_t ws_size,
                              hipStream_t stream)
{
    const float* feat   = (const float*)d_in[0];
    const int*   mask   = (const int*)  d_in[1];
    const int*   target = (const int*)  d_in[2];
    const float* startT = (const float*)d_in[3];
    const float* endT   = (const float*)d_in[4];
    const float* trans  = (const float*)d_in[5];
    float* out = (float*)d_out;

    zero_out<<<1, 256, 0, stream>>>(out, out_size);
    // 4 blocks x 512 threads = 64 scan waves, 16 waves/WGP -> 4 chains per SIMD
    crf_forward_scan<<<4, 512, 0, stream>>>(feat, mask, startT, endT, trans, out);
    // 128 blocks x 256 threads = 1024 numerator waves (one per batch row)
    crf_numerator<<<NB / 8, 256, 0, stream>>>(feat, mask, target, startT, endT, trans, out);
}